// DecodingLoss_BCEBased_80204219286147
// MI455X (gfx1250) — compile-verified
//
#include <hip/hip_runtime.h>

// Problem constants (fixed by the reference)
#define B_    128
#define T_    30
#define N_    10000
#define M_    5000
#define K_    10
#define CHK_W 6
#define OBS_W 50
#define EPSF  1e-6f

typedef __attribute__((ext_vector_type(2))) float v2f;
typedef __attribute__((ext_vector_type(8))) float v8f;

// gfx1250 has a native V_TANH_F32 TRANS instruction; use it when clang
// exposes the builtin (compile-safe fallback to libm tanhf otherwise).
#if defined(__has_builtin)
#if __has_builtin(__builtin_amdgcn_tanhf)
#define FAST_TANH(x) __builtin_amdgcn_tanhf(x)
#endif
#endif
#ifndef FAST_TANH
#define FAST_TANH(x) tanhf(x)
#endif

__global__ void init_ws_kernel(double* ws) { ws[0] = 0.0; }

__global__ void finalize_kernel(float* out, const double* ws) {
    out[0] = (float)(ws[0] * (1.0 / (double)(B_ * T_)));
}

__global__ __launch_bounds__(256)
void decoding_loss_kernel(const float* __restrict__ all_llrs,
                          const int*   __restrict__ syndromes,
                          const int*   __restrict__ observables,
                          const int*   __restrict__ chk_idx,
                          const int*   __restrict__ obs_idx,
                          double*      __restrict__ ws)
{
    __shared__ __align__(16) float tl[N_];   // tanh(0.5*llr) staged row (40 KB)
    __shared__ float wred[8];                // per-wave partial sums

    const int tid = threadIdx.x;
    const int bt  = blockIdx.x;              // (b, t) pair index, 0..B*T-1
    const int b   = bt / T_;

    // ---- Stage: load 10000 floats as float4, apply tanh once, store to LDS ----
    const float4* row = (const float4*)(all_llrs + (size_t)bt * N_);
    for (int i = tid; i < N_ / 4; i += 256) {
        float4 v = row[i];
        float4 t;
        t.x = FAST_TANH(0.5f * v.x);
        t.y = FAST_TANH(0.5f * v.y);
        t.z = FAST_TANH(0.5f * v.z);
        t.w = FAST_TANH(0.5f * v.w);
        ((float4*)tl)[i] = t;
    }
    __syncthreads();

    const float LOG2C = 0.69314718055994530942f;
    const float LO = -1.0f + EPSF, HI = 1.0f - EPSF;

    // ---- Check rows: BCE(-2*atanh(p), y) == log2 - log(1 + (1-2y)*p) ----
    float acc1 = 0.0f;
    const int* syn_b = syndromes + (size_t)b * M_;
    for (int m = tid; m < M_; m += 256) {
        const int* e = chk_idx + m * CHK_W;
        float p = tl[e[0]] * tl[e[1]];
        p *= tl[e[2]] * tl[e[3]];
        p *= tl[e[4]] * tl[e[5]];
        p = fminf(fmaxf(p, LO), HI);
        const float sgn = syn_b[m] ? -1.0f : 1.0f;
        acc1 += LOG2C - __logf(fmaf(sgn, p, 1.0f));   // v_log_f32 fast path
    }

    // ---- Observable rows (only 10, weight 50 each) ----
    float acc2 = 0.0f;
    if (tid < K_) {
        const int* e = obs_idx + tid * OBS_W;
        float p = 1.0f;
        #pragma unroll
        for (int j = 0; j < OBS_W; ++j) p *= tl[e[j]];
        p = fminf(fmaxf(p, LO), HI);
        const float sgn = observables[(size_t)b * K_ + tid] ? -1.0f : 1.0f;
        acc2 += LOG2C - __logf(fmaf(sgn, p, 1.0f));
    }

    float partial = 0.5f * acc1 + 0.5f * acc2;   // BETA = 0.5

    // ---- Wave-level reduction via V_WMMA_F32_16X16X4_F32 ----
    // A = all-ones 16x4  =>  D[m][n] = sum_k B[k][n] (column sums of B),
    // independent of the exact lane->element mapping. With B holding the 32
    // lane partials (second B VGPR = 0), the 16 column sums (duplicated in
    // lane halves) partition all 32 partials; summing D[.][0..15] over one
    // 16-lane half yields the exact f32 wave sum. EXEC is all-ones here
    // (uniform control flow across the block).
    v2f a   = { 1.0f, 1.0f };
    v2f bm  = { partial, 0.0f };
    v8f cz  = {};
    v8f d = __builtin_amdgcn_wmma_f32_16x16x4_f32(
        /*neg_a=*/false, a, /*neg_b=*/false, bm,
        /*c_mod=*/(short)0, cz, /*reuse_a=*/false, /*reuse_b=*/false);

    float cs = d[0];
    cs += __shfl_xor(cs, 8, 32);
    cs += __shfl_xor(cs, 4, 32);
    cs += __shfl_xor(cs, 2, 32);
    cs += __shfl_xor(cs, 1, 32);   // lanes 0..15 now hold the full wave sum

    if ((tid & 31) == 0) wred[tid >> 5] = cs;
    __syncthreads();

    if (tid == 0) {
        float s = 0.0f;
        #pragma unroll
        for (int w = 0; w < 8; ++w) s += wred[w];
        atomicAdd(ws, (double)s);   // fp64 accumulation across 3840 blocks
    }
}

extern "C" void kernel_launch(void* const* d_in, const int* in_sizes, int n_in,
                              void* d_out, int out_size, void* d_ws, size_t ws_size,
                              hipStream_t stream) {
    (void)in_sizes; (void)n_in; (void)out_size; (void)ws_size;
    const float* all_llrs    = (const float*)d_in[0];
    const int*   syndromes   = (const int*)d_in[1];
    const int*   observables = (const int*)d_in[2];
    const int*   chk_idx     = (const int*)d_in[3];
    // d_in[4] = chk_seg (implied by row-major layout; unused)
    const int*   obs_idx     = (const int*)d_in[5];
    // d_in[6] = obs_seg (unused)
    double* ws  = (double*)d_ws;
    float*  out = (float*)d_out;

    init_ws_kernel<<<1, 1, 0, stream>>>(ws);
    decoding_loss_kernel<<<B_ * T_, 256, 0, stream>>>(
        all_llrs, syndromes, observables, chk_idx, obs_idx, ws);
    finalize_kernel<<<1, 1, 0, stream>>>(out, ws);
}